// E55ScalarNoProjCell_47656957116865
// MI455X (gfx1250) — compile-verified
//
#include <hip/hip_runtime.h>
#include <math.h>

typedef __attribute__((ext_vector_type(2))) float v2f;
typedef __attribute__((ext_vector_type(8))) float v8f;

#define T_TOT   2048
#define BATCH   8
#define DMODEL  2048
#define NCH     (BATCH * DMODEL)   // 16384 independent channels
#define NCHUNK  32
#define CLEN    64                 // NCHUNK * CLEN == T_TOT
#define SUBS    (CLEN / 16)        // 16-step WMMA tiles per chunk

__device__ __forceinline__ float dev_sigmoid(float v) {
    return 1.0f / (1.0f + __expf(-v));
}

// fast sigmoid for the bulk elementwise path: v_exp_f32 + v_rcp_f32 (~1 ulp)
__device__ __forceinline__ float fast_sigmoid(float v) {
    return __builtin_amdgcn_rcpf(1.0f + __expf(-v));
}

// exact integer power by repeated multiply (e <= 16), once per wave at startup
__device__ __forceinline__ float powi(float base, int e) {
    float r = 1.0f;
    for (int i = 0; i < e; ++i) r *= base;
    return r;
}

// ---------------------------------------------------------------------------
// Pass 1: per (chunk, 4 channels) run the recurrence with zero initial state.
// End value is the chunk's additive contribution B_k:
//   h_end(chunk) = lam^CLEN * h_start(chunk) + B_k
// float4 per thread -> b128 loads; rows are 64 KiB apart so the unrolled body
// uses one base address + immediate offsets. Regular (RT) loads so x lands in
// the 192 MB L2 for pass 3's re-read.
// ---------------------------------------------------------------------------
__global__ void __launch_bounds__(256)
pass1_chunk_contrib(const float* __restrict__ x,
                    const float* __restrict__ loglam,
                    const float* __restrict__ bvec,
                    float* __restrict__ Bk) {
    const float lam = dev_sigmoid(loglam[0]);
    const int tid   = blockIdx.x * blockDim.x + threadIdx.x; // NCHUNK*NCH/4
    const int chunk = tid / (NCH / 4);
    const int c4    = tid % (NCH / 4);
    const int ch    = c4 * 4;                  // consecutive lanes -> coalesced

    const float4 bl = *reinterpret_cast<const float4*>(bvec + (ch % DMODEL));
    const float4* xp = reinterpret_cast<const float4*>(x) +
                       (chunk * CLEN * (NCH / 4) + c4);

    float4 h = make_float4(0.f, 0.f, 0.f, 0.f);
    for (int t0 = 0; t0 < CLEN; t0 += 8) {
#pragma unroll
        for (int u = 0; u < 8; ++u) {
            const float4 xv = xp[u * (NCH / 4)];   // immediate offsets u*64KiB
            h.x = fmaf(lam, xv.x + h.x, bl.x);
            h.y = fmaf(lam, xv.y + h.y, bl.y);
            h.z = fmaf(lam, xv.z + h.z, bl.z);
            h.w = fmaf(lam, xv.w + h.w, bl.w);
        }
        xp += 8 * (NCH / 4);
    }
    *reinterpret_cast<float4*>(Bk + chunk * NCH + ch) = h;
}

// ---------------------------------------------------------------------------
// Pass 2: serial prefix over the 32 chunk summaries (per 4 channels),
// plus h row 0 = h0.
// ---------------------------------------------------------------------------
__global__ void __launch_bounds__(256)
pass2_prefix(const float* __restrict__ loglam,
             const float* __restrict__ h0,
             const float* __restrict__ Bk,
             float* __restrict__ Hstart,
             float* __restrict__ h_out) {
    const float lam = dev_sigmoid(loglam[0]);
    float lamP = lam;                       // lam^CLEN = lam^64 by 6 squarings
#pragma unroll
    for (int i = 0; i < 6; ++i) lamP *= lamP;

    const int c4 = blockIdx.x * blockDim.x + threadIdx.x;   // NCH/4 threads
    const int ch = c4 * 4;

    float4 hs = *reinterpret_cast<const float4*>(h0 + ch);
    *reinterpret_cast<float4*>(h_out + ch) = hs;            // h[0] = h0
#pragma unroll
    for (int k = 0; k < NCHUNK; ++k) {
        *reinterpret_cast<float4*>(Hstart + k * NCH + ch) = hs;
        const float4 bk = *reinterpret_cast<const float4*>(Bk + k * NCH + ch);
        hs.x = fmaf(lamP, hs.x, bk.x);
        hs.y = fmaf(lamP, hs.y, bk.y);
        hs.z = fmaf(lamP, hs.z, bk.z);
        hs.w = fmaf(lamP, hs.w, bk.w);
    }
}

// ---------------------------------------------------------------------------
// Pass 3: one wave per (chunk, 16-channel tile). Each 16-step tile is
//   D[i][n] = sum_{j<=i} lam^{i-j} * u_j[n]  +  lam^{i+1} * hcarry[n]
// computed as 4 chained V_WMMA_F32_16X16X4_F32 (K = 16 split into 4).
//   A (16x4 f32): lane l -> M = l%16, K = v + 2*(l>>4)     (2 VGPRs)
//   B (4x16 f32): lane l -> N = l%16, K = v + 2*(l>>4)     (2 VGPRs)
//   C/D (16x16):  lane l -> N = l%16, M = r + 8*(l>>4)     (8 VGPRs)
// The hi-dependent row offsets are folded into per-lane base pointers once;
// all loads/stores inside the chunk then use constant immediate offsets
// (max ~4 MiB, within the signed 24-bit IOFFSET).
// x loads are non-temporal (read exactly once, from L2); out/h stores are
// non-temporal so the 256 MiB of streaming writes do not evict x from L2.
// Carry between 16-step tiles: D row M=15 (acc[7] in lanes 16..31) via one
// lane shuffle (ds_bpermute).
// ---------------------------------------------------------------------------
__global__ void __launch_bounds__(256)
pass3_wmma_scan(const float* __restrict__ x,
                const float* __restrict__ loglam,
                const float* __restrict__ bvec,
                const float* __restrict__ Hstart,
                float* __restrict__ out,
                float* __restrict__ h_out) {
    const float lam = dev_sigmoid(loglam[0]);

    const int lane  = threadIdx.x & 31;
    const int wave  = threadIdx.x >> 5;
    const int gwave = blockIdx.x * (blockDim.x >> 5) + wave;
    const int tilesPerChunk = NCH / 16;                 // 1024
    const int chunk = gwave / tilesPerChunk;
    const int tile  = gwave % tilesPerChunk;
    const int chan0 = tile * 16;
    const int hi    = lane >> 4;                        // 0 or 1
    const int lo    = lane & 15;

    // A = L, lower-triangular powers of lam (constant for the whole kernel)
    float Amat[4][2];
#pragma unroll
    for (int kb = 0; kb < 4; ++kb) {
#pragma unroll
        for (int v = 0; v < 2; ++v) {
            const int j    = 4 * kb + v + 2 * hi;       // K index 0..15
            const int diff = lo - j;                    // M - K
            Amat[kb][v] = (diff >= 0) ? powi(lam, diff) : 0.0f;
        }
    }
    // C scale: cpow[r] = lam^{(r + 8*hi) + 1}
    float cpow[8];
    {
        float p = powi(lam, 1 + 8 * hi);
#pragma unroll
        for (int r = 0; r < 8; ++r) { cpow[r] = p; p *= lam; }
    }

    const float bl = bvec[(chan0 % DMODEL) + lo];
    const int chbase = chan0 + lo;

    // Per-lane base pointers with hi folded in; everything else is an
    // immediate offset below.
    const float* xlane = x     + chunk * CLEN * NCH + chbase + 2 * hi * NCH;
    float*       hlane = h_out + (chunk * CLEN + 1) * NCH + chbase + 8 * hi * NCH;
    float*       olane = out   + chunk * CLEN * NCH + chbase + 8 * hi * NCH;

    float hcarry = Hstart[chunk * NCH + chan0 + lo];    // h_row[chunk*CLEN]

#pragma unroll
    for (int s = 0; s < SUBS; ++s) {
        v8f acc;
#pragma unroll
        for (int r = 0; r < 8; ++r) acc[r] = cpow[r] * hcarry;

#pragma unroll
        for (int kb = 0; kb < 4; ++kb) {
            v2f a; a.x = Amat[kb][0]; a.y = Amat[kb][1];
            v2f bm;
#pragma unroll
            for (int v = 0; v < 2; ++v) {
                // global j = 16*s + 4*kb + v + 2*hi; the 2*hi*NCH part is in
                // xlane, so the index here is a compile-time constant.
                const float xv =
                    __builtin_nontemporal_load(xlane + (16 * s + 4 * kb + v) * NCH);
                bm[v] = fmaf(lam, xv, bl);              // u_j = lam*x_j + b
            }
            acc = __builtin_amdgcn_wmma_f32_16x16x4_f32(
                false, a, false, bm, (short)0, acc, false, false);
        }

        // acc[r] = h_row[t0 + M + 1],  M = r + 8*hi (8*hi*NCH folded in base)
#pragma unroll
        for (int r = 0; r < 8; ++r) {
            const float hv = acc[r];
            __builtin_nontemporal_store(hv, hlane + (16 * s + r) * NCH);
            const float sg = fast_sigmoid(hv);          // v_exp + v_rcp
            __builtin_nontemporal_store(hv * hv * sg,   // h * silu(h)
                                        olane + (16 * s + r) * NCH);
        }

        // carry = h at M=15 (lanes 16..31 hold it in acc[7]); broadcast per N
        hcarry = __shfl(acc[7], 16 + lo, 32);
    }
}

// ---------------------------------------------------------------------------
extern "C" void kernel_launch(void* const* d_in, const int* in_sizes, int n_in,
                              void* d_out, int out_size, void* d_ws, size_t ws_size,
                              hipStream_t stream) {
    const float* x      = (const float*)d_in[0];   // [T, B, D]
    const float* loglam = (const float*)d_in[1];   // scalar
    const float* bvec   = (const float*)d_in[2];   // [D]
    const float* h0     = (const float*)d_in[3];   // [B, D]

    float* out   = (float*)d_out;                         // [T, B, D]
    float* h_out = out + (size_t)T_TOT * NCH;             // [T+1, B, D]

    float* Bk     = (float*)d_ws;                         // [NCHUNK, NCH]
    float* Hstart = Bk + (size_t)NCHUNK * NCH;            // [NCHUNK, NCH]

    pass1_chunk_contrib<<<(NCHUNK * NCH / 4) / 256, 256, 0, stream>>>(
        x, loglam, bvec, Bk);
    pass2_prefix<<<(NCH / 4) / 256, 256, 0, stream>>>(
        loglam, h0, Bk, Hstart, h_out);
    pass3_wmma_scan<<<(NCHUNK * (NCH / 16)) / 8, 256, 0, stream>>>(
        x, loglam, bvec, Hstart, out, h_out);
}